// GCN_1073741824392
// MI455X (gfx1250) — compile-verified
//
#include <hip/hip_runtime.h>
#include <math.h>

#define NODES 100000
#define FEATS 128
#define NF (NODES * FEATS)
#define RED_BLOCKS 1024

typedef float v2f __attribute__((ext_vector_type(2)));
typedef float v8f __attribute__((ext_vector_type(8)));

// ---------------- utility: zero a float buffer ----------------
__global__ void zero_kernel(float* __restrict__ p, int n) {
    int i = blockIdx.x * blockDim.x + threadIdx.x;
    int stride = gridDim.x * blockDim.x;
    for (; i < n; i += stride) p[i] = 0.0f;
}

// ---------------- degree counting (exact integer float atomics) ----------------
__global__ void degree_kernel(const int* __restrict__ src, const int* __restrict__ dst,
                              float* __restrict__ dsrc, float* __restrict__ ddst, int E) {
    int e = blockIdx.x * blockDim.x + threadIdx.x;
    if (e < E) {
        atomicAdd(&dsrc[src[e]], 1.0f);
        atomicAdd(&ddst[dst[e]], 1.0f);
    }
}

// ---------------- deg -> 1/sqrt(max(deg,1)) in place ----------------
__global__ void norm_kernel(float* __restrict__ a, float* __restrict__ b, int n) {
    int i = blockIdx.x * blockDim.x + threadIdx.x;
    if (i < n) {
        a[i] = 1.0f / sqrtf(fmaxf(a[i], 1.0f));
        b[i] = 1.0f / sqrtf(fmaxf(b[i], 1.0f));
    }
}

// ---------------- SpMM: m[dst] += h[src] * out_norm[src]  (one wave32 per edge) ----------------
__global__ __launch_bounds__(256) void spmm_kernel(const float* __restrict__ h,
                                                   const float* __restrict__ onorm,
                                                   const int* __restrict__ src,
                                                   const int* __restrict__ dst,
                                                   float* __restrict__ m, int E) {
    int gid = blockIdx.x * blockDim.x + threadIdx.x;
    int e = gid >> 5;              // wave id == edge id
    int lane = threadIdx.x & 31;   // lane handles 4 consecutive feats
    if (e >= E) return;
    int s = src[e];
    int d = dst[e];
    float w = onorm[s];
    const float4* hp = (const float4*)(h + (long)s * FEATS);
    float4 v = hp[lane];
    float* mp = m + (long)d * FEATS + lane * 4;
    atomicAdd(mp + 0, v.x * w);
    atomicAdd(mp + 1, v.y * w);
    atomicAdd(mp + 2, v.z * w);
    atomicAdd(mp + 3, v.w * w);
}

// ---------------- fused (m * in_norm) @ W + bias, ReLU via V_WMMA_F32_16X16X4_F32 ----------------
// One wave computes a 16-row stripe x 128 output cols (8 tiles of 16x16).
__global__ __launch_bounds__(128) void gemm_relu_kernel(const float* __restrict__ X,
                                                        const float* __restrict__ inorm,
                                                        const float* __restrict__ W,
                                                        const float* __restrict__ bias,
                                                        float* __restrict__ Y, int nrows) {
    int wave = threadIdx.x >> 5;
    int lane = threadIdx.x & 31;
    int l16  = lane & 15;
    int hi   = lane >> 4;          // which half of the wave (K-split for A/B frags)
    int stripe = blockIdx.x * 4 + wave;
    int mrow = stripe * 16;
    if (mrow >= nrows) return;     // wave-uniform: EXEC stays all-ones for active waves

    v8f acc[8];
    v8f zeroacc = {0.f, 0.f, 0.f, 0.f, 0.f, 0.f, 0.f, 0.f};
#pragma unroll
    for (int n = 0; n < 8; n++) acc[n] = zeroacc;

    float inn = inorm[mrow + l16];
    const float* xrow = X + (long)(mrow + l16) * FEATS;

    for (int kb = 0; kb < FEATS; kb += 4) {
        int k0 = kb + 2 * hi;
        // A frag (16x4 f32): lanes 0-15 -> K=kb,kb+1 ; lanes 16-31 -> K=kb+2,kb+3 ; M = l16
        v2f a;
        a.x = xrow[k0]     * inn;
        a.y = xrow[k0 + 1] * inn;
        const float* w0 = W + (long)k0 * FEATS;
        const float* w1 = w0 + FEATS;
#pragma unroll
        for (int n = 0; n < 8; n++) {
            // B frag (4x16 f32): V0=row k0, V1=row k0+1, N = l16 within tile n
            v2f b;
            b.x = w0[n * 16 + l16];
            b.y = w1[n * 16 + l16];
            acc[n] = __builtin_amdgcn_wmma_f32_16x16x4_f32(
                false, a, false, b, (short)0, acc[n], false, false);
        }
    }

#pragma unroll
    for (int n = 0; n < 8; n++) {
        float bn = bias[n * 16 + l16];
#pragma unroll
        for (int r = 0; r < 8; r++) {
            int row = mrow + r + 8 * hi;   // C/D layout: VGPR r -> M = r (lanes 0-15), r+8 (16-31)
            float v = acc[n][r] + bn;
            Y[(long)row * FEATS + n * 16 + l16] = v > 0.0f ? v : 0.0f;
        }
    }
}

// ---------------- deterministic Frobenius-norm reduction ----------------
__global__ __launch_bounds__(256) void sumsq_kernel(const float* __restrict__ x, int n,
                                                    float* __restrict__ partials) {
    __shared__ float sm[256];
    int tid = threadIdx.x;
    int i = blockIdx.x * 256 + tid;
    const int stride = RED_BLOCKS * 256;
    float s = 0.0f;
    for (; i < n; i += stride) {
        float v = x[i];
        s += v * v;
    }
    sm[tid] = s;
    __syncthreads();
    for (int off = 128; off > 0; off >>= 1) {
        if (tid < off) sm[tid] += sm[tid + off];
        __syncthreads();
    }
    if (tid == 0) partials[blockIdx.x] = sm[0];
}

__global__ __launch_bounds__(256) void finish_kernel(const float* __restrict__ partials,
                                                     float* __restrict__ total) {
    __shared__ float sm[256];
    int tid = threadIdx.x;
    float s = 0.0f;
    for (int i = tid; i < RED_BLOCKS; i += 256) s += partials[i];
    sm[tid] = s;
    __syncthreads();
    for (int off = 128; off > 0; off >>= 1) {
        if (tid < off) sm[tid] += sm[tid + off];
        __syncthreads();
    }
    if (tid == 0) total[0] = sm[0];
}

__global__ void scale_kernel(const float* __restrict__ x, const float* __restrict__ total,
                             float* __restrict__ y, int n) {
    int i = blockIdx.x * blockDim.x + threadIdx.x;
    int stride = gridDim.x * blockDim.x;
    float s = 1.0f / sqrtf(total[0]);
    for (; i < n; i += stride) y[i] = x[i] * s;
}

extern "C" void kernel_launch(void* const* d_in, const int* in_sizes, int n_in,
                              void* d_out, int out_size, void* d_ws, size_t ws_size,
                              hipStream_t stream) {
    const float* emb = (const float*)d_in[0];
    const float* W0  = (const float*)d_in[1];
    const float* b0  = (const float*)d_in[2];
    const float* W1  = (const float*)d_in[3];
    const float* b1  = (const float*)d_in[4];
    const float* W2  = (const float*)d_in[5];
    const float* b2  = (const float*)d_in[6];
    // d_in[7] = input_nodes : arange(N) identity gather -> unused
    const int* src = (const int*)d_in[8];
    const int* dst = (const int*)d_in[9];
    const int E = in_sizes[8];

    float* out = (float*)d_out;
    float* ws  = (float*)d_ws;

    float* m        = ws;                         // [N,128] aggregation buffer
    float* hA       = ws + (size_t)NF;            // [N,128] ping buffer
    float* onorm    = ws + 2 * (size_t)NF;        // [N] src-degree -> out_norm
    float* inorm    = onorm + NODES;              // [N] dst-degree -> in_norm
    float* partials = inorm + NODES;              // [RED_BLOCKS]
    float* total    = partials + RED_BLOCKS;      // [1]

    const int gemmBlocks = (NODES / 16 + 3) / 4;  // 4 waves/block, 16 rows/wave
    const int spmmBlocks = (E + 7) / 8;           // 8 waves/block, 1 edge/wave

    // degrees + norms
    zero_kernel<<<512, 256, 0, stream>>>(onorm, 2 * NODES);
    degree_kernel<<<(E + 255) / 256, 256, 0, stream>>>(src, dst, onorm, inorm, E);
    norm_kernel<<<(NODES + 255) / 256, 256, 0, stream>>>(onorm, inorm, NODES);

    // layer 0: emb -> hA
    zero_kernel<<<4096, 256, 0, stream>>>(m, NF);
    spmm_kernel<<<spmmBlocks, 256, 0, stream>>>(emb, onorm, src, dst, m, E);
    gemm_relu_kernel<<<gemmBlocks, 128, 0, stream>>>(m, inorm, W0, b0, hA, NODES);

    // layer 1: hA -> out (ping-pong through d_out)
    zero_kernel<<<4096, 256, 0, stream>>>(m, NF);
    spmm_kernel<<<spmmBlocks, 256, 0, stream>>>(hA, onorm, src, dst, m, E);
    gemm_relu_kernel<<<gemmBlocks, 128, 0, stream>>>(m, inorm, W1, b1, out, NODES);

    // layer 2: out -> hA
    zero_kernel<<<4096, 256, 0, stream>>>(m, NF);
    spmm_kernel<<<spmmBlocks, 256, 0, stream>>>(out, onorm, src, dst, m, E);
    gemm_relu_kernel<<<gemmBlocks, 128, 0, stream>>>(m, inorm, W2, b2, hA, NODES);

    // global Frobenius-norm normalize: out = hA / ||hA||
    sumsq_kernel<<<RED_BLOCKS, 256, 0, stream>>>(hA, NF, partials);
    finish_kernel<<<1, 256, 0, stream>>>(partials, total);
    scale_kernel<<<2048, 256, 0, stream>>>(hA, total, out, NF);
}